// GIN_64527588655342
// MI455X (gfx1250) — compile-verified
//
#include <hip/hip_runtime.h>

#define NODES   100000
#define EDGES   1600000
#define DDIM    32
#define NGRAPH  128
#define EPW     8          // edges per wave in scatter kernels

typedef float v2f __attribute__((ext_vector_type(2)));
typedef float v8f __attribute__((ext_vector_type(8)));

// ---------------------------------------------------------------------------
// K0: agg1 = feats (folds the "+x" of GINConv), out = 0
// ---------------------------------------------------------------------------
__global__ void gin_init(const float4* __restrict__ feats4,
                         float4* __restrict__ agg4,
                         float* __restrict__ out, int n4) {
  int i = blockIdx.x * blockDim.x + threadIdx.x;
  if (i < n4)            agg4[i] = feats4[i];
  if (i < NGRAPH * DDIM) out[i]  = 0.0f;
}

// ---------------------------------------------------------------------------
// K1/K3: agg[dst] += X[src].  One edge per wave iteration, lane = feature.
// Every gather read and every atomic hits exactly one 128B L2 line.
// ---------------------------------------------------------------------------
__global__ void gin_edge_scatter(const float* __restrict__ X,
                                 const int* __restrict__ src,
                                 const int* __restrict__ dst,
                                 float* __restrict__ agg) {
  int wave = (blockIdx.x * blockDim.x + threadIdx.x) >> 5;
  int lane = threadIdx.x & 31;
  long e0 = (long)wave * EPW;
  for (int i = 0; i < EPW; ++i) {
    long e = e0 + i;                    // uniform across the wave
    if (e >= EDGES) break;              // uniform exit, EXEC stays full
    int s = src[e];                     // scalarized broadcast loads
    int d = dst[e];
    float v = X[(size_t)s * DDIM + lane];
    atomicAdd(&agg[(size_t)d * DDIM + lane], v);   // global_atomic_add_f32
  }
}

// ---------------------------------------------------------------------------
// WMMA core: one wave computes a 16x32 tile of  H = act(X @ W + b).
// A (16x4 f32): lane L holds row L&15, K-pair 2*(L>>4)   -> v2f
// B (4x16 f32): lane L holds col L&15, K-pair 2*(L>>4)   -> v2f
// C (16x16 f32): lane L -> n = L&15, m = v + 8*(L>>4)    -> v8f
// K=32 chained as 8 x (16x16x4) per accumulator, 2 accumulators (N=0..15,16..31)
// ---------------------------------------------------------------------------
__device__ __forceinline__ void gin_tile_compute(
    const float* __restrict__ X, const float* __restrict__ W,
    const float* __restrict__ bvec, int row0, int lane,
    v8f& c0, v8f& c1) {
  int n    = lane & 15;          // output column (and A-row for this lane)
  int koff = (lane >> 4) * 2;    // 0 or 2: which K-pair this half-wave owns
  const float* xrow = X + (size_t)(row0 + n) * DDIM;

  float bb0 = bvec[n], bb1 = bvec[n + 16];
#pragma unroll
  for (int v = 0; v < 8; ++v) { c0[v] = bb0; c1[v] = bb1; }

#pragma unroll
  for (int kc = 0; kc < DDIM; kc += 4) {
    v2f a, bA, bB;
    a[0]  = xrow[kc + koff];
    a[1]  = xrow[kc + koff + 1];
    bA[0] = W[(kc + koff)     * DDIM + n];
    bA[1] = W[(kc + koff + 1) * DDIM + n];
    bB[0] = W[(kc + koff)     * DDIM + 16 + n];
    bB[1] = W[(kc + koff + 1) * DDIM + 16 + n];
    c0 = __builtin_amdgcn_wmma_f32_16x16x4_f32(false, a, false, bA,
                                               (short)0, c0, false, false);
    c1 = __builtin_amdgcn_wmma_f32_16x16x4_f32(false, a, false, bB,
                                               (short)0, c1, false, false);
  }
}

// K2: H = relu(X @ W + b); also mirrors H into H2 (initializes agg2 = h1)
__global__ void gin_wmma_layer(const float* __restrict__ X,
                               const float* __restrict__ W,
                               const float* __restrict__ bvec,
                               float* __restrict__ H,
                               float* __restrict__ H2) {
  int wave = (blockIdx.x * blockDim.x + threadIdx.x) >> 5;
  int lane = threadIdx.x & 31;
  int row0 = wave * 16;
  if (row0 >= NODES) return;           // whole-wave uniform exit

  v8f c0, c1;
  gin_tile_compute(X, W, bvec, row0, lane, c0, c1);

  int n     = lane & 15;
  int mbase = row0 + 8 * (lane >> 4);
#pragma unroll
  for (int v = 0; v < 8; ++v) {
    size_t r = (size_t)(mbase + v) * DDIM;
    float y0 = fmaxf(c0[v], 0.0f);
    float y1 = fmaxf(c1[v], 0.0f);
    H [r + n]      = y0;  H [r + 16 + n] = y1;
    H2[r + n]      = y0;  H2[r + 16 + n] = y1;
  }
}

// K4: layer-2 GEMM fused with SumPooling: out[gid[m]] += (X @ W + b)[m]
__global__ void gin_wmma_pool(const float* __restrict__ X,
                              const float* __restrict__ W,
                              const float* __restrict__ bvec,
                              const int* __restrict__ gid,
                              float* __restrict__ out) {
  int wave = (blockIdx.x * blockDim.x + threadIdx.x) >> 5;
  int lane = threadIdx.x & 31;
  int row0 = wave * 16;
  if (row0 >= NODES) return;

  v8f c0, c1;
  gin_tile_compute(X, W, bvec, row0, lane, c0, c1);

  int n     = lane & 15;
  int mbase = row0 + 8 * (lane >> 4);
#pragma unroll
  for (int v = 0; v < 8; ++v) {
    int g = gid[mbase + v];
    atomicAdd(&out[(size_t)g * DDIM + n],      c0[v]);
    atomicAdd(&out[(size_t)g * DDIM + 16 + n], c1[v]);
  }
}

// ---------------------------------------------------------------------------
extern "C" void kernel_launch(void* const* d_in, const int* in_sizes, int n_in,
                              void* d_out, int out_size, void* d_ws, size_t ws_size,
                              hipStream_t stream) {
  const float* feats = (const float*)d_in[0];
  const int*   src   = (const int*)  d_in[1];
  const int*   dst   = (const int*)  d_in[2];
  const int*   gid   = (const int*)  d_in[3];
  const float* W1    = (const float*)d_in[4];
  const float* b1    = (const float*)d_in[5];
  const float* W2    = (const float*)d_in[6];
  const float* b2    = (const float*)d_in[7];
  float* out = (float*)d_out;

  const size_t NF = (size_t)NODES * DDIM;          // 3.2M floats per buffer
  float* agg1 = (float*)d_ws;
  float* h1   = agg1 + NF;
  float* agg2 = h1   + NF;

  // K0: agg1 = feats; out = 0
  {
    int n4 = (int)(NF / 4);
    gin_init<<<(n4 + 255) / 256, 256, 0, stream>>>(
        (const float4*)feats, (float4*)agg1, out, n4);
  }

  // K1: agg1 += scatter(feats)
  {
    long waves = (EDGES + EPW - 1) / EPW;          // 200000 waves
    long thr   = waves * 32;
    gin_edge_scatter<<<(int)((thr + 255) / 256), 256, 0, stream>>>(
        feats, src, dst, agg1);
  }

  // K2: h1 = relu(agg1 @ W1 + b1); agg2 = h1
  {
    long tiles = NODES / 16;                       // 6250 (exact)
    long thr   = tiles * 32;
    gin_wmma_layer<<<(int)((thr + 255) / 256), 256, 0, stream>>>(
        agg1, W1, b1, h1, agg2);
  }

  // K3: agg2 += scatter(h1)
  {
    long waves = (EDGES + EPW - 1) / EPW;
    long thr   = waves * 32;
    gin_edge_scatter<<<(int)((thr + 255) / 256), 256, 0, stream>>>(
        h1, src, dst, agg2);
  }

  // K4: out[g] += (agg2 @ W2 + b2)
  {
    long tiles = NODES / 16;
    long thr   = tiles * 32;
    gin_wmma_pool<<<(int)((thr + 255) / 256), 256, 0, stream>>>(
        agg2, W2, b2, gid, out);
  }
}